// GINE_68573447848159
// MI455X (gfx1250) — compile-verified
//
#include <hip/hip_runtime.h>

typedef __attribute__((ext_vector_type(2))) float v2f;
typedef __attribute__((ext_vector_type(8))) float v8f;
typedef __attribute__((ext_vector_type(4))) unsigned int u32x4;
typedef __attribute__((ext_vector_type(8))) int i32x8;
typedef __attribute__((ext_vector_type(4))) int i32x4;

constexpr int kNodes  = 50000;
constexpr int kEdges  = 800000;
constexpr int kFN     = 32;
constexpr int kFE     = 16;
constexpr int kHid    = 128;
constexpr int kDepth  = 4;
constexpr int kGraphs = 128;
constexpr int kNT     = kHid / 16;    // 8 column tiles of 16
constexpr int kStP    = 132;          // padded LDS row stride (bank-conflict free)

__device__ __forceinline__ v8f wmma_f32(v2f a, v2f b, v8f c) {
  // V_WMMA_F32_16X16X4_F32 : D = A(16x4) * B(4x16) + C(16x16), fp32 native
  return __builtin_amdgcn_wmma_f32_16x16x4_f32(false, a, false, b, (short)0, c,
                                               false, false);
}

// TDM: DMA a 64KB contiguous block (8192 x 8B elements, single row tile) from
// global memory into LDS at byte offset lds_off. Descriptor per CDNA5 ISA
// CDNA5 ISA. Wave-level op; tracked by TENSORcnt.
__device__ __forceinline__ void tdm_load_64k(const void* gptr, unsigned lds_off) {
  unsigned long long ga = (unsigned long long)(uintptr_t)gptr;
  u32x4 g0;
  g0[0] = 1u;                                          // count=1 (valid), user mode
  g0[1] = lds_off;                                     // lds_addr (bytes)
  g0[2] = (unsigned)ga;                                // global_addr[31:0]
  g0[3] = (unsigned)((ga >> 32) & 0x1FFFFFFu)          // global_addr[56:32]
          | (2u << 30);                                // type=2 ("image")
  i32x8 g1;
  g1[0] = (int)(3u << 16);                             // data_size=3 (8 bytes)
  g1[1] = (int)((8192u & 0xFFFFu) << 16);              // tensor_dim0[15:0]=8192
  g1[2] = (int)((8192u >> 16) | (1u << 16));           // dim0[31:16] | tensor_dim1=1
  g1[3] = (int)(8192u << 16);                          // dim1[31:16]=0 | tile_dim0=8192
  g1[4] = 1;                                           // tile_dim1=1, tile_dim2=0
  g1[5] = 8192;                                        // tensor_dim0_stride[31:0]
  g1[6] = (int)((8192u & 0xFFFFu) << 16);              // stride0[47:32]=0 | stride1[15:0]
  g1[7] = 0;                                           // stride1[47:16]=0
  i32x4 gz = {0, 0, 0, 0};
#if defined(__clang_major__) && __clang_major__ >= 23
  i32x8 gz8 = {0, 0, 0, 0, 0, 0, 0, 0};
  __builtin_amdgcn_tensor_load_to_lds(g0, g1, gz, gz, gz8, 0);
#else
  __builtin_amdgcn_tensor_load_to_lds(g0, g1, gz, gz, 0);
#endif
}

__global__ void k_zero(float* __restrict__ p, int n4) {
  int i = blockIdx.x * blockDim.x + threadIdx.x;
  if (i < n4) ((float4*)p)[i] = make_float4(0.f, 0.f, 0.f, 0.f);
}

// Repack row-major W[K][128] into B-fragment order:
// WP2[(t*8+nt)*32+lane] = { W[4t+kb][16nt+nn], W[4t+kb+1][16nt+nn] },
// kb = 2*(lane>>4), nn = lane&15  ->  one coalesced b64 load per WMMA.
__global__ void k_pack(const float* __restrict__ w, float* __restrict__ wp, int K) {
  int i = blockIdx.x * blockDim.x + threadIdx.x;
  int total = (K / 4) * kNT * 32;
  if (i >= total) return;
  int lane = i & 31;
  int nt   = (i >> 5) & 7;
  int t    = i >> 8;
  int kb   = (lane >> 4) << 1;
  int nn   = lane & 15;
  int krow = t * 4 + kb;
  float2 v;
  v.x = w[(size_t)krow * kHid + nt * 16 + nn];
  v.y = w[(size_t)(krow + 1) * kHid + nt * 16 + nn];
  ((float2*)wp)[i] = v;
}

// h = relu(x @ init_w + init_b); also writes h0 copy. One 16-row tile per wave.
__global__ void __launch_bounds__(128) k_init(const float* __restrict__ x,
                                              const float* __restrict__ wp,
                                              const float* __restrict__ bias,
                                              float* __restrict__ h,
                                              float* __restrict__ h0) {
  const int lane = threadIdx.x & 31;
  const int wave = (blockIdx.x * blockDim.x + threadIdx.x) >> 5;
  if (wave >= kNodes / 16) return;
  const int mrow = (wave << 4) + (lane & 15);
  const int kb   = (lane >> 4) << 1;
  const int nn   = lane & 15;
  const float2* bf = (const float2*)wp;
  v2f a[kFN / 4];
#pragma unroll
  for (int t = 0; t < kFN / 4; ++t) {
    const float* p = x + (size_t)mrow * kFN + t * 4 + kb;
    a[t].x = p[0]; a[t].y = p[1];
  }
  const int rbase = (wave << 4) + ((lane >> 4) << 3);
#pragma unroll
  for (int nt = 0; nt < kNT; ++nt) {
    v8f c = {};
#pragma unroll
    for (int t = 0; t < kFN / 4; ++t) {
      float2 bb = bf[(t * 8 + nt) * 32 + lane];
      v2f b; b.x = bb.x; b.y = bb.y;
      c = wmma_f32(a[t], b, c);
    }
    const float bv = bias[nt * 16 + nn];
#pragma unroll
    for (int i = 0; i < 8; ++i) {
      float v = fmaxf(c[i] + bv, 0.0f);
      size_t off = (size_t)(rbase + i) * kHid + nt * 16 + nn;
      h[off] = v; h0[off] = v;
    }
  }
}

// Fused GINE message: m = relu(h[src] + edge_attr @ We + be); atomic scatter to aggr[dst].
__global__ void __launch_bounds__(128) k_edge(const float* __restrict__ ea,
                                              const long long* __restrict__ srcIdx,
                                              const long long* __restrict__ dstIdx,
                                              const float* __restrict__ wp,
                                              const float* __restrict__ be,
                                              const float* __restrict__ h,
                                              float* __restrict__ aggr) {
  const int lane = threadIdx.x & 31;
  const int wave = (blockIdx.x * blockDim.x + threadIdx.x) >> 5;
  if (wave >= kEdges / 16) return;
  const int erow = (wave << 4) + (lane & 15);
  const int kb   = (lane >> 4) << 1;
  const int nn   = lane & 15;
  if (wave + 1 < kEdges / 16)
    __builtin_prefetch(ea + (size_t)(erow + 16) * kFE, 0, 0);  // global_prefetch_b8
  const float2* bf = (const float2*)wp;
  v2f a[kFE / 4];
#pragma unroll
  for (int t = 0; t < kFE / 4; ++t) {
    const float* p = ea + (size_t)erow * kFE + t * 4 + kb;
    a[t].x = p[0]; a[t].y = p[1];
  }
  const int rbase = (wave << 4) + ((lane >> 4) << 3);
  int sr[8], dr[8];
#pragma unroll
  for (int i = 0; i < 8; ++i) {
    sr[i] = (int)srcIdx[rbase + i];
    dr[i] = (int)dstIdx[rbase + i];
  }
#pragma unroll
  for (int nt = 0; nt < kNT; ++nt) {
    v8f c = {};
#pragma unroll
    for (int t = 0; t < kFE / 4; ++t) {
      float2 bb = bf[(t * 8 + nt) * 32 + lane];
      v2f b; b.x = bb.x; b.y = bb.y;
      c = wmma_f32(a[t], b, c);
    }
    const float bv = be[nt * 16 + nn];
    const int col = nt * 16 + nn;
#pragma unroll
    for (int i = 0; i < 8; ++i) {
      float v = fmaxf(c[i] + bv + h[(size_t)sr[i] * kHid + col], 0.0f);
      unsafeAtomicAdd(&aggr[(size_t)dr[i] * kHid + col], v);   // global_atomic_add_f32
    }
  }
}

// z = h + aggr; hout = relu( relu(z@W1+b1)@W2 + b2 + h0 ).
// Packed W1/W2 staged in LDS once per block via TDM (tensor_load_to_lds);
// grid-stride over row tiles; intermediate staged in wave-private padded LDS.
__global__ void __launch_bounds__(128) k_mlp(const float* __restrict__ h,
                                             const float* __restrict__ aggr,
                                             const float* __restrict__ h0,
                                             const float* __restrict__ wp1,
                                             const float* __restrict__ b1,
                                             const float* __restrict__ wp2,
                                             const float* __restrict__ b2,
                                             float* __restrict__ hout) {
  extern __shared__ float lds[];
  float* s_w1 = lds;              // 16384 floats (64 KB) @ LDS offset 0
  float* s_w2 = lds + 16384;      // 16384 floats (64 KB) @ LDS offset 65536
  const int lane = threadIdx.x & 31;
  const int wl   = threadIdx.x >> 5;
  float* st = lds + 32768 + wl * (16 * kStP);   // 16 x 132 per wave

  if (threadIdx.x < 32) {          // wave 0 issues the DMA (wave-level ops)
    tdm_load_64k(wp1, 0u);
    tdm_load_64k(wp2, 65536u);
    __builtin_amdgcn_s_wait_tensorcnt((short)0);   // s_wait_tensorcnt 0
  }
  __syncthreads();                 // s_barrier_signal / s_barrier_wait

  const int kb   = (lane >> 4) << 1;
  const int nn   = lane & 15;
  const int roff = (lane >> 4) << 3;
  const float2* bf1 = (const float2*)s_w1;
  const float2* bf2 = (const float2*)s_w2;

  float bv1[kNT], bv2[kNT];
#pragma unroll
  for (int nt = 0; nt < kNT; ++nt) {
    bv1[nt] = b1[nt * 16 + nn];
    bv2[nt] = b2[nt * 16 + nn];
  }

  const int ntiles = kNodes / 16;
  const int stride = gridDim.x * 4;
  for (int tile = blockIdx.x * 4 + wl; tile < ntiles; tile += stride) {
    const int mrow = (tile << 4) + nn;
    const int nxt = tile + stride;
    if (nxt < ntiles) {
      __builtin_prefetch(h    + (size_t)((nxt << 4) + nn) * kHid, 0, 0);
      __builtin_prefetch(aggr + (size_t)((nxt << 4) + nn) * kHid, 0, 0);
    }
    // ---- GEMM1: A = z = h + aggr ----
    v2f a[kHid / 4];
#pragma unroll
    for (int t = 0; t < kHid / 4; ++t) {
      size_t off = (size_t)mrow * kHid + t * 4 + kb;
      a[t].x = h[off] + aggr[off];
      a[t].y = h[off + 1] + aggr[off + 1];
    }
#pragma unroll
    for (int nt = 0; nt < kNT; ++nt) {
      v8f c = {};
#pragma unroll
      for (int t = 0; t < kHid / 4; ++t) {
        float2 bb = bf1[(t * 8 + nt) * 32 + lane];   // ds_load_b64
        v2f b; b.x = bb.x; b.y = bb.y;
        c = wmma_f32(a[t], b, c);
      }
#pragma unroll
      for (int i = 0; i < 8; ++i)
        st[(roff + i) * kStP + nt * 16 + nn] = fmaxf(c[i] + bv1[nt], 0.0f);
    }
    // ---- GEMM2: A from wave-private LDS (in-order DS, conflict-free pads) ----
#pragma unroll
    for (int t = 0; t < kHid / 4; ++t) {
      const float* p = st + nn * kStP + t * 4 + kb;
      a[t].x = p[0]; a[t].y = p[1];                  // ds_load_b64
    }
#pragma unroll
    for (int nt = 0; nt < kNT; ++nt) {
      v8f c = {};
#pragma unroll
      for (int t = 0; t < kHid / 4; ++t) {
        float2 bb = bf2[(t * 8 + nt) * 32 + lane];
        v2f b; b.x = bb.x; b.y = bb.y;
        c = wmma_f32(a[t], b, c);
      }
      const int col = nt * 16 + nn;
#pragma unroll
      for (int i = 0; i < 8; ++i) {
        size_t off = (size_t)((tile << 4) + roff + i) * kHid + col;
        hout[off] = fmaxf(c[i] + bv2[nt] + h0[off], 0.0f);
      }
    }
  }
}

// global_add_pool over sorted batch ids: run-length accumulate, atomic flush.
__global__ void __launch_bounds__(128) k_pool(const float* __restrict__ h,
                                              const long long* __restrict__ batch,
                                              float* __restrict__ pooled) {
  const int c = threadIdx.x;
  int n0 = blockIdx.x * 250;
  if (n0 >= kNodes) return;
  int n1 = n0 + 250; if (n1 > kNodes) n1 = kNodes;
  long long g = batch[n0];
  float s = 0.0f;
  for (int n = n0; n < n1; ++n) {
    long long gg = batch[n];
    if (gg != g) {
      unsafeAtomicAdd(&pooled[(size_t)g * kHid + c], s);
      s = 0.0f; g = gg;
    }
    s += h[(size_t)n * kHid + c];
  }
  unsafeAtomicAdd(&pooled[(size_t)g * kHid + c], s);
}

__global__ void k_final(const float* __restrict__ pooled,
                        const float* __restrict__ fw,
                        const float* __restrict__ fb,
                        float* __restrict__ out) {
  int g = blockIdx.x * blockDim.x + threadIdx.x;
  if (g >= kGraphs) return;
  float s = fb[0];
  for (int k = 0; k < kHid; ++k) s += pooled[(size_t)g * kHid + k] * fw[k];
  out[g] = s;
}

extern "C" void kernel_launch(void* const* d_in, const int* in_sizes, int n_in,
                              void* d_out, int out_size, void* d_ws, size_t ws_size,
                              hipStream_t stream) {
  const float*     x      = (const float*)d_in[0];
  const long long* ei     = (const long long*)d_in[1];   // int64 [2, E]
  const float*     ea     = (const float*)d_in[2];
  const long long* batch  = (const long long*)d_in[3];   // int64 [N]
  const float*     init_w = (const float*)d_in[4];
  const float*     init_b = (const float*)d_in[5];
  const float*     edge_w = (const float*)d_in[6];
  const float*     edge_b = (const float*)d_in[7];
  const float*     mlp_w1 = (const float*)d_in[8];
  const float*     mlp_b1 = (const float*)d_in[9];
  const float*     mlp_w2 = (const float*)d_in[10];
  const float*     mlp_b2 = (const float*)d_in[11];
  const float*     ffn_w  = (const float*)d_in[12];
  const float*     ffn_b  = (const float*)d_in[13];
  (void)in_sizes; (void)n_in; (void)out_size; (void)ws_size;

  const size_t nelems = (size_t)kNodes * kHid;
  char* ws = (char*)d_ws;
  float* h_a     = (float*)ws; ws += nelems * sizeof(float);
  float* h_b     = (float*)ws; ws += nelems * sizeof(float);
  float* h0      = (float*)ws; ws += nelems * sizeof(float);
  float* aggr    = (float*)ws; ws += nelems * sizeof(float);
  float* pooled  = (float*)ws; ws += (size_t)kGraphs * kHid * sizeof(float);
  float* pk_init = (float*)ws; ws += (size_t)kFN * kHid * sizeof(float);
  float* pk_edge = (float*)ws; ws += (size_t)kDepth * kFE * kHid * sizeof(float);
  float* pk_w1   = (float*)ws; ws += (size_t)kDepth * kHid * kHid * sizeof(float);
  float* pk_w2   = (float*)ws;

  // ---- one-shot weight repacking into B-fragment order ----
  k_pack<<<(kFN * 64 + 255) / 256, 256, 0, stream>>>(init_w, pk_init, kFN);
  for (int d = 0; d < kDepth; ++d) {
    k_pack<<<(kFE * 64 + 255) / 256, 256, 0, stream>>>(
        edge_w + (size_t)d * kFE * kHid, pk_edge + (size_t)d * kFE * kHid, kFE);
    k_pack<<<(kHid * 64 + 255) / 256, 256, 0, stream>>>(
        mlp_w1 + (size_t)d * kHid * kHid, pk_w1 + (size_t)d * kHid * kHid, kHid);
    k_pack<<<(kHid * 64 + 255) / 256, 256, 0, stream>>>(
        mlp_w2 + (size_t)d * kHid * kHid, pk_w2 + (size_t)d * kHid * kHid, kHid);
  }

  const int waves_node = kNodes / 16;   // 3125
  const int waves_edge = kEdges / 16;   // 50000
  k_init<<<(waves_node + 3) / 4, 128, 0, stream>>>(x, pk_init, init_b, h_a, h0);

  const long long* srcI = ei;
  const long long* dstI = ei + kEdges;
  float* hcur = h_a; float* hnext = h_b;
  const int n4 = (int)(nelems / 4);
  const size_t mlp_lds = (32768 + 4 * 16 * kStP) * sizeof(float);   // ~161 KB
  for (int d = 0; d < kDepth; ++d) {
    k_zero<<<(n4 + 255) / 256, 256, 0, stream>>>(aggr, n4);
    k_edge<<<(waves_edge + 3) / 4, 128, 0, stream>>>(
        ea, srcI, dstI, pk_edge + (size_t)d * kFE * kHid,
        edge_b + (size_t)d * kHid, hcur, aggr);
    k_mlp<<<256, 128, mlp_lds, stream>>>(
        hcur, aggr, h0, pk_w1 + (size_t)d * kHid * kHid,
        mlp_b1 + (size_t)d * kHid, pk_w2 + (size_t)d * kHid * kHid,
        mlp_b2 + (size_t)d * kHid, hnext);
    float* t = hcur; hcur = hnext; hnext = t;
  }
  const int pz = kGraphs * kHid / 4;
  k_zero<<<(pz + 255) / 256, 256, 0, stream>>>(pooled, pz);
  k_pool<<<(kNodes + 249) / 250, 128, 0, stream>>>(hcur, batch, pooled);
  k_final<<<1, 128, 0, stream>>>(pooled, ffn_w, ffn_b, (float*)d_out);
}